// PatchAggregator_32710470927139
// MI455X (gfx1250) — compile-verified
//
#include <hip/hip_runtime.h>

#ifndef __has_builtin
#define __has_builtin(x) 0
#endif

#if __has_builtin(__builtin_amdgcn_global_load_async_to_lds_b128) && \
    __has_builtin(__builtin_amdgcn_s_wait_asynccnt)
#define USE_ASYNC_LDS 1
#else
#define USE_ASYNC_LDS 0
#endif

#define BB 8
#define KK 256
#define CC 8
#define PS 64
#define HH 512
#define WW 512
#define HWPIX (HH * WW)          // 262144 = 2^18
#define PATCH_PIX (PS * PS)      // 4096
#define NTHREADS 256
#define FILL_VALUE -10.0f

// Builtin signature (from hipcc diagnostic): params are int4* with AS quals.
typedef int v4i __attribute__((ext_vector_type(4)));
typedef __attribute__((address_space(1))) v4i* gptr_v4i;
typedef __attribute__((address_space(3))) v4i* lptr_v4i;

// Copy one 64x64 f32 channel tile (16 KB) global -> LDS with 256 threads.
// 16 B per lane per op -> 4 async instructions per wave per tile.
__device__ __forceinline__ void tile_copy_async(const float* __restrict__ gsrc,
                                                float* __restrict__ ldst,
                                                int tid) {
#if USE_ASYNC_LDS
#pragma unroll
  for (int q = 0; q < 4; ++q) {
    int e = (tid + q * NTHREADS) * 4;  // float index, 16B granules
    float* g = const_cast<float*>(gsrc + e);
    __builtin_amdgcn_global_load_async_to_lds_b128(
        (gptr_v4i)(v4i*)g, (lptr_v4i)(v4i*)(ldst + e), /*offset=*/0,
        /*cpol=*/0);
  }
#else
#pragma unroll
  for (int q = 0; q < 4; ++q) {
    int e = (tid + q * NTHREADS) * 4;
    *(float4*)(ldst + e) = *(const float4*)(gsrc + e);
  }
#endif
}

__device__ __forceinline__ void wait_async_le(int pending) {
#if USE_ASYNC_LDS
  if (pending == 4)
    __builtin_amdgcn_s_wait_asynccnt(4);
  else
    __builtin_amdgcn_s_wait_asynccnt(0);
#else
  (void)pending;
#endif
}

__global__ __launch_bounds__(NTHREADS) void patchagg_zero_kernel(
    float* __restrict__ sum, unsigned* __restrict__ cnt) {
  size_t idx = (size_t)blockIdx.x * NTHREADS + threadIdx.x;
  if (idx < (size_t)BB * CC * HWPIX) sum[idx] = 0.0f;
  if (idx < (size_t)BB * HWPIX) cnt[idx] = 0u;
}

__global__ __launch_bounds__(NTHREADS) void patchagg_scatter_kernel(
    const float* __restrict__ patch_logits,  // [B,K,C,64,64]
    const int* __restrict__ coords,          // [B,K,2] (int32 on device)
    float* __restrict__ sum,                 // [B,C,H,W] accumulators (d_out)
    unsigned* __restrict__ cnt) {            // [B,H,W] coverage counts (d_ws)
  __shared__ float buf[2][PATCH_PIX];  // 32 KB double buffer

  const int blk = blockIdx.x;   // b*K + k
  const int b = blk >> 8;       // K == 256
  const int tid = threadIdx.x;

  const int r0 = coords[blk * 2 + 0];
  const int c0 = coords[blk * 2 + 1];

  const float* src = patch_logits + (size_t)blk * (CC * PATCH_PIX);

  // Kick off DMA of channel 0 immediately.
  tile_copy_async(src, buf[0], tid);

  // Coverage counts do not need the LDS data -> overlap with the first DMA.
  unsigned* cntb = cnt + (size_t)b * HWPIX;
#pragma unroll
  for (int q = 0; q < PATCH_PIX / NTHREADS; ++q) {
    const int p = tid + q * NTHREADS;
    const int row = r0 + (p >> 6);
    const int col = c0 + (p & 63);
    if ((unsigned)row < HH && (unsigned)col < WW)
      atomicAdd(&cntb[row * WW + col], 1u);
  }

  // Pipelined per-channel scatter: DMA of channel c+1 overlaps atomics of c.
  for (int c = 0; c < CC; ++c) {
    if (c + 1 < CC)
      tile_copy_async(src + (size_t)(c + 1) * PATCH_PIX, buf[(c + 1) & 1], tid);
    wait_async_le((c + 1 < CC) ? 4 : 0);  // oldest 4 (channel c) retired
    __syncthreads();                      // channel c visible to all waves

    float* dst = sum + ((size_t)b * CC + c) * HWPIX;
    const float* lb = buf[c & 1];
#pragma unroll
    for (int q = 0; q < PATCH_PIX / NTHREADS; ++q) {
      const int p = tid + q * NTHREADS;
      const int row = r0 + (p >> 6);
      const int col = c0 + (p & 63);
      if ((unsigned)row < HH && (unsigned)col < WW)
        atomicAdd(&dst[row * WW + col], lb[p]);
    }
    __syncthreads();  // protect buf[c&1] before it is re-targeted at iter c+1
  }
}

__global__ __launch_bounds__(NTHREADS) void patchagg_finalize_kernel(
    float* __restrict__ out, const unsigned* __restrict__ cnt) {
  const size_t idx = (size_t)blockIdx.x * NTHREADS + threadIdx.x;
  if (idx >= (size_t)BB * CC * HWPIX) return;
  const size_t pixel = idx & (HWPIX - 1);  // HW = 2^18
  const size_t b = idx >> 21;              // C*HW = 2^21
  const unsigned n = cnt[(b << 18) + pixel];
  const float s = out[idx];
  // n is an exact integer count: n > MIN_COVERAGE <=> n >= 1, max(n,eps)=n.
  out[idx] = (n != 0u) ? (s / (float)n) : FILL_VALUE;
}

extern "C" void kernel_launch(void* const* d_in, const int* in_sizes, int n_in,
                              void* d_out, int out_size, void* d_ws,
                              size_t ws_size, hipStream_t stream) {
  const float* patch_logits = (const float*)d_in[0];
  const int* coords = (const int*)d_in[1];  // JAX x64 disabled -> int32
  float* out = (float*)d_out;               // doubles as the sum accumulator
  unsigned* cnt = (unsigned*)d_ws;          // 8 MB of workspace

  const int total_out = BB * CC * HWPIX;    // 16,777,216
  const int nblk = total_out / NTHREADS;    // 65536

  patchagg_zero_kernel<<<nblk, NTHREADS, 0, stream>>>(out, cnt);
  patchagg_scatter_kernel<<<BB * KK, NTHREADS, 0, stream>>>(patch_logits,
                                                            coords, out, cnt);
  patchagg_finalize_kernel<<<nblk, NTHREADS, 0, stream>>>(out, cnt);
}